// Pointnet2MSG_46222438040238
// MI455X (gfx1250) — compile-verified
//
#include <hip/hip_runtime.h>
#include <hip/hip_bf16.h>

typedef __attribute__((ext_vector_type(2))) float v2f;
typedef __attribute__((ext_vector_type(8))) float v8f;

// ---------------------------------------------------------------------------
// Extract xyz (first 3 ch) and feat (last 6 ch) from (B, N, 9) pointcloud
// ---------------------------------------------------------------------------
__global__ void extract_kernel(const float* __restrict__ pc,
                               float* __restrict__ xyz, float* __restrict__ feat,
                               int total /*B*N*/) {
    int t = blockIdx.x * blockDim.x + threadIdx.x;
    if (t >= total) return;
    const float* p = pc + (size_t)t * 9;
    xyz[t * 3 + 0] = p[0];
    xyz[t * 3 + 1] = p[1];
    xyz[t * 3 + 2] = p[2];
#pragma unroll
    for (int c = 0; c < 6; ++c) feat[(size_t)t * 6 + c] = p[3 + c];
}

// ---------------------------------------------------------------------------
// Farthest point sampling: one block per batch, dists live in LDS (N<=8192)
// ---------------------------------------------------------------------------
#define FPS_THREADS 512
__global__ void fps_kernel(const float* __restrict__ xyz, int N, int npoint,
                           int* __restrict__ idx_out) {
    __shared__ float D[8192];
    __shared__ float s_val[FPS_THREADS];
    __shared__ int   s_idx[FPS_THREADS];
    __shared__ int   s_sel;
    int b = blockIdx.x;
    const float* X = xyz + (size_t)b * N * 3;
    int tid = threadIdx.x;
    for (int i = tid; i < N; i += FPS_THREADS) D[i] = 1e10f;
    if (tid == 0) s_sel = 0;
    __syncthreads();
    for (int it = 0; it < npoint; ++it) {
        int sel = s_sel;
        if (tid == 0) idx_out[(size_t)b * npoint + it] = sel;
        float cx = X[sel * 3 + 0], cy = X[sel * 3 + 1], cz = X[sel * 3 + 2];
        float best = -1.0f;
        int   bi = 0x7fffffff;
        for (int i = tid; i < N; i += FPS_THREADS) {
            float dx = X[i * 3 + 0] - cx;
            float dy = X[i * 3 + 1] - cy;
            float dz = X[i * 3 + 2] - cz;
            float d = dx * dx + dy * dy + dz * dz;
            float dd = D[i];
            if (d < dd) { dd = d; D[i] = dd; }
            if (dd > best) { best = dd; bi = i; }
        }
        s_val[tid] = best;
        s_idx[tid] = bi;
        __syncthreads();
        for (int s = FPS_THREADS / 2; s > 0; s >>= 1) {
            if (tid < s) {
                float ov = s_val[tid + s];
                int   oi = s_idx[tid + s];
                if (ov > s_val[tid] || (ov == s_val[tid] && oi < s_idx[tid])) {
                    s_val[tid] = ov;
                    s_idx[tid] = oi;
                }
            }
            __syncthreads();
        }
        if (tid == 0) s_sel = s_idx[0];
        __syncthreads();
    }
}

// ---------------------------------------------------------------------------
// Gather sampled centers' coordinates
// ---------------------------------------------------------------------------
__global__ void gather_xyz_kernel(const float* __restrict__ xyz, const int* __restrict__ idx,
                                  float* __restrict__ out, int N, int npoint, int total) {
    int t = blockIdx.x * blockDim.x + threadIdx.x;
    if (t >= total) return;
    int b = t / npoint;
    int p = idx[t];
    const float* src = xyz + ((size_t)b * N + p) * 3;
    out[t * 3 + 0] = src[0];
    out[t * 3 + 1] = src[1];
    out[t * 3 + 2] = src[2];
}

// ---------------------------------------------------------------------------
// Ball query: first K indices (ascending) within radius, pad with first found
// ---------------------------------------------------------------------------
__global__ void ball_query_kernel(const float* __restrict__ xyz, const float* __restrict__ new_xyz,
                                  int* __restrict__ idx, int N, int npoint, int K, float r2,
                                  int total /*B*npoint*/) {
    int t = blockIdx.x * blockDim.x + threadIdx.x;
    if (t >= total) return;
    int b = t / npoint;
    const float* X = xyz + (size_t)b * N * 3;
    float cx = new_xyz[t * 3 + 0], cy = new_xyz[t * 3 + 1], cz = new_xyz[t * 3 + 2];
    int* out = idx + (size_t)t * K;
    int cnt = 0;
    int first = 0;
    for (int i = 0; i < N && cnt < K; ++i) {
        float dx = X[i * 3 + 0] - cx;
        float dy = X[i * 3 + 1] - cy;
        float dz = X[i * 3 + 2] - cz;
        float d = dx * dx + dy * dy + dz * dz;
        if (d < r2) {
            if (cnt == 0) first = i;
            out[cnt++] = i;
        }
    }
    for (; cnt < K; ++cnt) out[cnt] = first;
}

// ---------------------------------------------------------------------------
// Build grouped rows: [xyz - center | feat]  ->  (B*npoint*K, 3+Cf)
// ---------------------------------------------------------------------------
__global__ void group_concat_kernel(const float* __restrict__ xyz, const float* __restrict__ feat,
                                    const float* __restrict__ new_xyz, const int* __restrict__ idx,
                                    float* __restrict__ out, int N, int npoint, int K, int Cf,
                                    long total) {
    int C = 3 + Cf;
    long stride = (long)gridDim.x * blockDim.x;
    for (long t = (long)blockIdx.x * blockDim.x + threadIdx.x; t < total; t += stride) {
        int  c = (int)(t % C);
        long rowid = t / C;        // b*npoint*K + s*K + k
        long bs = rowid / K;       // b*npoint + s
        int  b = (int)(bs / npoint);
        int  p = idx[rowid];
        float v;
        if (c < 3) v = xyz[((size_t)b * N + p) * 3 + c] - new_xyz[bs * 3 + c];
        else       v = feat[((size_t)b * N + p) * Cf + (c - 3)];
        out[t] = v;
    }
}

// ---------------------------------------------------------------------------
// WMMA fp32 GEMM + ReLU:  Out[M, Nout] = relu(A[M,Kd] @ W[Nout,Kd]^T)
// Requirements used by launch: M % 128 == 0, Nout % 16 == 0 (true for every
// layer in this network).  Block = 8 waves; block tile = 128 x 16.
// Both operands staged in LDS with k-tail zero padding, so the inner 64-deep
// K block is 16 fully-unrolled back-to-back V_WMMA_F32_16X16X4_F32 fed purely
// by ds_load -- no predication, EXEC stays all-ones around the WMMAs.
// ---------------------------------------------------------------------------
#define GEMM_WAVES 8
#define KTILE 64
#define ASTRIDE (KTILE + 1)   // pad LDS A rows to 65 floats: conflict-free frag reads
__global__ void gemm_relu_wmma(const float* __restrict__ A, const float* __restrict__ W,
                               float* __restrict__ Out, int M, int Kd, int Nout) {
    __shared__ float At[128 * ASTRIDE];   // 33,280 B
    __shared__ float Wt[KTILE * 16];      //  4,096 B
    int tid = threadIdx.x;
    int lane = tid & 31;
    int wave = tid >> 5;
    int rlane = lane & 15;
    int khalf = (lane >> 4) * 2;          // 0 or 2
    int mBase = blockIdx.x * 128;         // always < M (M % 128 == 0)
    int n0 = blockIdx.y * 16;             // always < Nout (Nout % 16 == 0)
    int rowLocal = wave * 16 + rlane;     // 0..127
    v8f acc = {};
    for (int k0 = 0; k0 < Kd; k0 += KTILE) {
        __syncthreads();
        // --- W tile: W[n0..n0+15, k0..k0+63] -> Wt[k][n], zero-padded in k ---
        for (int t = tid; t < KTILE * 16; t += 256) {
            int kk = t >> 4;
            int nn = t & 15;
            int gk = k0 + kk;
            int ck = gk < Kd ? gk : Kd - 1;                 // clamped: always in-range
            float v = W[(size_t)(n0 + nn) * Kd + ck];
            Wt[t] = (gk < Kd) ? v : 0.0f;
        }
        // --- A tile: A[mBase..+127, k0..k0+63] -> At, zero-padded in k ---
        for (int t = tid; t < 128 * KTILE; t += 256) {
            int r = t >> 6;                                 // 0..127
            int kk = t & 63;
            int gk = k0 + kk;
            int ck = gk < Kd ? gk : 0;                      // clamped: always in-range
            float v = A[(size_t)(mBase + r) * Kd + ck];
            At[r * ASTRIDE + kk] = (gk < Kd) ? v : 0.0f;
        }
        __syncthreads();
        if (k0 + KTILE < Kd)
            __builtin_prefetch(&A[(size_t)(mBase + rowLocal) * Kd + k0 + KTILE], 0, 1);
        const float* arow = &At[rowLocal * ASTRIDE];
        int kend = Kd - k0;
        if (kend >= KTILE) {
            // full tile: 16 back-to-back WMMAs
#pragma unroll
            for (int kk = 0; kk < KTILE; kk += 4) {
                v2f a, b;
                a.x = arow[kk + khalf];
                a.y = arow[kk + khalf + 1];
                b.x = Wt[(kk + khalf) * 16 + rlane];
                b.y = Wt[(kk + khalf + 1) * 16 + rlane];
                acc = __builtin_amdgcn_wmma_f32_16x16x4_f32(
                    false, a, false, b, (short)0, acc, false, false);
            }
        } else {
            // K tail (block-uniform trip count; zero-padded LDS covers partial group)
            for (int kk = 0; kk < kend; kk += 4) {
                v2f a, b;
                a.x = arow[kk + khalf];
                a.y = arow[kk + khalf + 1];
                b.x = Wt[(kk + khalf) * 16 + rlane];
                b.y = Wt[(kk + khalf + 1) * 16 + rlane];
                acc = __builtin_amdgcn_wmma_f32_16x16x4_f32(
                    false, a, false, b, (short)0, acc, false, false);
            }
        }
    }
    int outRow = mBase + wave * 16 + (lane >> 4) * 8;
#pragma unroll
    for (int i = 0; i < 8; ++i) {
        float v = acc[i];
        v = v > 0.0f ? v : 0.0f;                            // ReLU (every layer)
        Out[(size_t)(outRow + i) * Nout + n0 + rlane] = v;
    }
}

// ---------------------------------------------------------------------------
// Max pool over K samples; write at channel offset within strided feature row
// ---------------------------------------------------------------------------
__global__ void maxpool_kernel(const float* __restrict__ hid, float* __restrict__ out,
                               int npoint, int K, int C, int outStride, int outOff,
                               int total /*B*npoint*C*/) {
    int t = blockIdx.x * blockDim.x + threadIdx.x;
    if (t >= total) return;
    int c = t % C;
    int bs = t / C;
    const float* h = hid + ((size_t)bs * K) * C + c;
    float m = h[0];
    for (int k = 1; k < K; ++k) {
        float v = h[(size_t)k * C];
        m = v > m ? v : m;
    }
    out[(size_t)bs * outStride + outOff + c] = m;
}

// ---------------------------------------------------------------------------
// 3-NN search + inverse-distance weights
// ---------------------------------------------------------------------------
__global__ void knn3_kernel(const float* __restrict__ xu, const float* __restrict__ xk,
                            int* __restrict__ idx3, float* __restrict__ w3,
                            int Nunk, int Nkn, int total /*B*Nunk*/) {
    int t = blockIdx.x * blockDim.x + threadIdx.x;
    if (t >= total) return;
    int b = t / Nunk;
    float px = xu[t * 3 + 0], py = xu[t * 3 + 1], pz = xu[t * 3 + 2];
    const float* X = xk + (size_t)b * Nkn * 3;
    float d0 = 1e30f, d1 = 1e30f, d2 = 1e30f;
    int   i0 = 0, i1 = 0, i2 = 0;
    for (int i = 0; i < Nkn; ++i) {
        float dx = X[i * 3 + 0] - px;
        float dy = X[i * 3 + 1] - py;
        float dz = X[i * 3 + 2] - pz;
        float d = dx * dx + dy * dy + dz * dz;
        if (d < d0)      { d2 = d1; i2 = i1; d1 = d0; i1 = i0; d0 = d; i0 = i; }
        else if (d < d1) { d2 = d1; i2 = i1; d1 = d; i1 = i; }
        else if (d < d2) { d2 = d; i2 = i; }
    }
    float w0 = 1.0f / (d0 + 1e-8f);
    float w1 = 1.0f / (d1 + 1e-8f);
    float w2 = 1.0f / (d2 + 1e-8f);
    float s = w0 + w1 + w2;
    idx3[t * 3 + 0] = i0; idx3[t * 3 + 1] = i1; idx3[t * 3 + 2] = i2;
    w3[t * 3 + 0] = w0 / s; w3[t * 3 + 1] = w1 / s; w3[t * 3 + 2] = w2 / s;
}

// ---------------------------------------------------------------------------
// Interpolate known features + concat unknown features -> (B*Nunk, Ck+Cu)
// ---------------------------------------------------------------------------
__global__ void interp_concat_kernel(const float* __restrict__ fk, const float* __restrict__ fu,
                                     const int* __restrict__ idx3, const float* __restrict__ w3,
                                     float* __restrict__ out, int Nunk, int Nkn, int Ck, int Cu,
                                     long total) {
    int C = Ck + Cu;
    long stride = (long)gridDim.x * blockDim.x;
    for (long t = (long)blockIdx.x * blockDim.x + threadIdx.x; t < total; t += stride) {
        int  c = (int)(t % C);
        long n = t / C;            // b*Nunk + row
        int  b = (int)(n / Nunk);
        float v;
        if (c < Ck) {
            const float* F = fk + (size_t)b * Nkn * Ck;
            v = w3[n * 3 + 0] * F[(size_t)idx3[n * 3 + 0] * Ck + c] +
                w3[n * 3 + 1] * F[(size_t)idx3[n * 3 + 1] * Ck + c] +
                w3[n * 3 + 2] * F[(size_t)idx3[n * 3 + 2] * Ck + c];
        } else {
            v = fu[n * (long)Cu + (c - Ck)];
        }
        out[t] = v;
    }
}

// ---------------------------------------------------------------------------
// (B, N, C) -> (B, C, N)
// ---------------------------------------------------------------------------
__global__ void transpose_out_kernel(const float* __restrict__ f, float* __restrict__ out,
                                     int N, int C, int total) {
    int t = blockIdx.x * blockDim.x + threadIdx.x;
    if (t >= total) return;
    int  c = t % C;
    long n = t / C;
    int  b = (int)(n / N);
    int  nn = (int)(n % N);
    out[((size_t)b * C + c) * N + nn] = f[t];
}

// ---------------------------------------------------------------------------
// Host orchestration
// ---------------------------------------------------------------------------
static inline void launch_gemm(const float* A, const float* W, float* Out,
                               int M, int Kd, int Nout, hipStream_t s) {
    dim3 grid(M / 128, Nout / 16);        // M % 128 == 0, Nout % 16 == 0 by construction
    gemm_relu_wmma<<<grid, 32 * GEMM_WAVES, 0, s>>>(A, W, Out, M, Kd, Nout);
}

extern "C" void kernel_launch(void* const* d_in, const int* in_sizes, int n_in,
                              void* d_out, int out_size, void* d_ws, size_t ws_size,
                              hipStream_t stream) {
    (void)in_sizes; (void)n_in; (void)out_size; (void)ws_size;
    const int B = 2, N0 = 8192;
    const float* pc = (const float*)d_in[0];

    // weights in setup_inputs() insertion order: sa[4][2][2], then fp[4][2]
    const float* saW[4][2][2];
    const float* fpW[4][2];
    int di = 1;
    for (int l = 0; l < 4; ++l)
        for (int sc = 0; sc < 2; ++sc)
            for (int ly = 0; ly < 2; ++ly) saW[l][sc][ly] = (const float*)d_in[di++];
    for (int l = 0; l < 4; ++l)
        for (int ly = 0; ly < 2; ++ly) fpW[l][ly] = (const float*)d_in[di++];

    // bump-allocate workspace (floats)
    float* wsf = (float*)d_ws;
    size_t off = 0;
    auto alloc = [&](size_t n) { float* p = wsf + off; off += n; return p; };

    float* xyz0  = alloc((size_t)B * N0 * 3);
    float* feat0 = alloc((size_t)B * N0 * 6);
    int NP[5] = {N0, 4096, 1024, 256, 64};
    float* xyzL[5];
    xyzL[0] = xyz0;
    for (int l = 1; l < 5; ++l) xyzL[l] = alloc((size_t)B * NP[l] * 3);
    float* feat1  = alloc((size_t)B * 4096 * 256);
    float* feat2  = alloc((size_t)B * 1024 * 512);
    float* feat3  = alloc((size_t)B * 256 * 512);
    float* feat4  = alloc((size_t)B * 64 * 128);
    float* feat0n = alloc((size_t)B * N0 * 32);
    int* fps_idx  = (int*)alloc((size_t)B * 4096);
    int* ball_idx = (int*)alloc((size_t)B * 4096 * 32);
    int* knn_idx  = (int*)alloc((size_t)B * N0 * 3);
    float* knn_w  = alloc((size_t)B * N0 * 3);
    float* catbuf = alloc((size_t)4587520);   // max concat rows*cols
    float* hidA   = alloc((size_t)8388608);   // layer-1 activations
    float* hidB   = alloc((size_t)8388608);   // layer-2 activations (SA)

    // ---- extract xyz/feat from pointcloud ----
    {
        int total = B * N0;
        extract_kernel<<<(total + 255) / 256, 256, 0, stream>>>(pc, xyz0, feat0, total);
    }

    // ---- SA levels ----
    int   Nin_[4]  = {8192, 4096, 1024, 256};
    int   np_[4]   = {4096, 1024, 256, 64};
    int   Cf_[4]   = {6, 48, 64, 128};
    float rad_[4][2] = {{0.1f, 0.5f}, {0.5f, 1.0f}, {1.0f, 2.0f}, {2.0f, 4.0f}};
    int   Ks_[2]   = {16, 32};
    int   Ch_[4][2] = {{16, 32}, {32, 32}, {64, 64}, {64, 64}};
    int   Ctot_[4] = {48, 64, 128, 128};
    float* featIn[4]  = {feat0, feat1, feat2, feat3};
    float* featOut[4] = {feat1, feat2, feat3, feat4};

    for (int l = 0; l < 4; ++l) {
        int Nin = Nin_[l], np = np_[l], Cf = Cf_[l];
        fps_kernel<<<B, FPS_THREADS, 0, stream>>>(xyzL[l], Nin, np, fps_idx);
        int tg = B * np;
        gather_xyz_kernel<<<(tg + 255) / 256, 256, 0, stream>>>(xyzL[l], fps_idx, xyzL[l + 1],
                                                                Nin, np, tg);
        int choff = 0;
        for (int sc = 0; sc < 2; ++sc) {
            int K = Ks_[sc];
            float r = rad_[l][sc];
            ball_query_kernel<<<(tg + 127) / 128, 128, 0, stream>>>(
                xyzL[l], xyzL[l + 1], ball_idx, Nin, np, K, r * r, tg);
            int C = 3 + Cf;
            long totg = (long)B * np * K * C;
            int gb = (int)((totg + 255) / 256);
            if (gb > 16384) gb = 16384;
            group_concat_kernel<<<gb, 256, 0, stream>>>(
                xyzL[l], featIn[l], xyzL[l + 1], ball_idx, catbuf, Nin, np, K, Cf, totg);
            int M = B * np * K;
            int C1 = Ch_[l][sc];
            launch_gemm(catbuf, saW[l][sc][0], hidA, M, C, C1, stream);
            launch_gemm(hidA,   saW[l][sc][1], hidB, M, C1, C1, stream);
            int tp = B * np * C1;
            maxpool_kernel<<<(tp + 255) / 256, 256, 0, stream>>>(
                hidB, featOut[l], np, K, C1, Ctot_[l], choff, tp);
            choff += C1;
        }
    }

    // ---- FP levels (in reference order: i = -1 .. -4) ----
    int Nunk_[4] = {256, 1024, 4096, 8192};
    int Nkn_[4]  = {64, 256, 1024, 4096};
    int Ckn_[4]  = {128, 512, 512, 256};
    int Cunk_[4] = {128, 64, 48, 6};
    int Cho_[4][2] = {{512, 512}, {512, 512}, {256, 256}, {32, 32}};
    float* fk_[4] = {feat4, feat3, feat2, feat1};
    float* fu_[4] = {feat3, feat2, feat1, feat0};
    float* fo_[4] = {feat3, feat2, feat1, feat0n};
    float* xu_[4] = {xyzL[3], xyzL[2], xyzL[1], xyzL[0]};
    float* xk_[4] = {xyzL[4], xyzL[3], xyzL[2], xyzL[1]};
    int fpIdx_[4] = {3, 2, 1, 0};

    for (int s = 0; s < 4; ++s) {
        int Nu = Nunk_[s], Nk = Nkn_[s];
        int tt = B * Nu;
        knn3_kernel<<<(tt + 127) / 128, 128, 0, stream>>>(xu_[s], xk_[s], knn_idx, knn_w,
                                                          Nu, Nk, tt);
        int Ck = Ckn_[s], Cu = Cunk_[s], C = Ck + Cu;
        long totc = (long)B * Nu * C;
        int gb = (int)((totc + 255) / 256);
        if (gb > 16384) gb = 16384;
        interp_concat_kernel<<<gb, 256, 0, stream>>>(fk_[s], fu_[s], knn_idx, knn_w,
                                                     catbuf, Nu, Nk, Ck, Cu, totc);
        int M = B * Nu;
        int H = Cho_[s][0], O = Cho_[s][1];
        launch_gemm(catbuf, fpW[fpIdx_[s]][0], hidA, M, C, H, stream);
        launch_gemm(hidA,   fpW[fpIdx_[s]][1], fo_[s], M, H, O, stream);
    }

    // ---- final transpose (B, N, 32) -> (B, 32, N) ----
    int to = B * N0 * 32;
    transpose_out_kernel<<<(to + 255) / 256, 256, 0, stream>>>(feat0n, (float*)d_out,
                                                               N0, 32, to);
}